// SANVQA_19250043421102
// MI455X (gfx1250) — compile-verified
//
#include <hip/hip_runtime.h>
#include <hip/hip_bf16.h>

typedef __attribute__((ext_vector_type(2))) float v2f;
typedef __attribute__((ext_vector_type(4))) float v4f;
typedef __attribute__((ext_vector_type(8))) float v8f;

#define DIM 300
#define BATCH 8
#define NE 30
#define IMG 224
#define HW (IMG*IMG)          // 50176 = 196*256 = 49*1024
#define NT_N 19               // ceil(300/16)
#define DCH 75                // d-chunk for wordgrid kernel (300/4)

// ---------------- GEMM: Y = relu(X @ W^T + bias), via V_WMMA_F32_16X16X4_F32 ----
// X: [R, 300], W: [300, 300] (row n of W dotted with row of X), Y: [R, 300]
// Loads are UNCONDITIONAL (row pointers clamped to row 0, which always exists and
// is finite); out-of-range lanes are zeroed with a register select after the load
// so the inner loop has no exec-mask manipulation around the WMMAs.
__global__ void linear_relu_wmma(const float* __restrict__ X,
                                 const float* __restrict__ W,
                                 const float* __restrict__ bias,
                                 float* __restrict__ Y,
                                 int R, int ntiles)
{
    const int wave = threadIdx.x >> 5;
    const int tile = blockIdx.x * (blockDim.x >> 5) + wave;
    if (tile >= ntiles) return;                 // wave-uniform branch
    const int mt = tile / NT_N, nt = tile % NT_N;
    const int m_base = mt * 16, n_base = nt * 16;
    const int lane = threadIdx.x & 31;
    const int half = lane >> 4;                 // 0: lanes 0-15, 1: lanes 16-31
    const int l15  = lane & 15;
    const int m = m_base + l15;
    const int n = n_base + l15;
    const bool mv = (m < R);
    const bool nv = (n < DIM);
    const float* xrow = X + (size_t)(mv ? m : 0) * DIM;   // clamped -> always valid
    const float* wrow = W + (size_t)(nv ? n : 0) * DIM;

    const v2f zero2 = {0.f, 0.f};
    v8f c = {0.f,0.f,0.f,0.f,0.f,0.f,0.f,0.f};
#pragma unroll 5
    for (int k0 = 0; k0 < DIM; k0 += 4) {
        const int k = k0 + 2 * half;            // 8B-aligned (k even, row stride even)
        // A 16x4 (ISA 7.12.2): lane holds row m, K = k0+2*half+{0,1}  -> b64 load
        v2f a = *reinterpret_cast<const v2f*>(xrow + k);
        // B 4x16: row-striped across lanes; VGPR v holds K = 2*half+v, col n
        v2f b = *reinterpret_cast<const v2f*>(wrow + k);
        a = mv ? a : zero2;                     // v_cndmask, no branch
        b = nv ? b : zero2;
        c = __builtin_amdgcn_wmma_f32_16x16x4_f32(false, a, false, b,
                                                  (short)0, c, false, false);
    }
    const float bn = nv ? bias[n] : 0.f;
#pragma unroll
    for (int r = 0; r < 8; ++r) {
        const int row = m_base + r + 8*half;    // C/D: VGPR r -> M=r (lo half), M=r+8 (hi half)
        if (row < R && nv) {
            float v = c[r] + bn;
            Y[(size_t)row * DIM + n] = v > 0.f ? v : 0.f;
        }
    }
}

// ---------------- per-row L2 normalize (in place), one block (128 thr) per row ---
__global__ void rownorm_kernel(float* __restrict__ Y)
{
    float* p = Y + (size_t)blockIdx.x * DIM;
    const int tid = threadIdx.x;
    float s = 0.f;
    for (int d = tid; d < DIM; d += 128) { float v = p[d]; s += v * v; }
    for (int off = 16; off; off >>= 1) s += __shfl_down(s, off);
    __shared__ float red[4];
    if ((tid & 31) == 0) red[tid >> 5] = s;
    __syncthreads();
    const float tot = red[0] + red[1] + red[2] + red[3];
    const float den = fmaxf(sqrtf(tot), 1e-12f);
    for (int d = tid; d < DIM; d += 128) p[d] = p[d] / den;
}

// ---------------- scores[b,e] = qn[b] . embn[b,e]; one wave per (b,e) -----------
__global__ void score_kernel(const float* __restrict__ embn,
                             const float* __restrict__ qn,
                             float* __restrict__ scores)
{
    const int wave = threadIdx.x >> 5;
    const int p = blockIdx.x * (blockDim.x >> 5) + wave;
    if (p >= BATCH * NE) return;
    const int b = p / NE;
    const int lane = threadIdx.x & 31;
    const float* er = embn + (size_t)p * DIM;
    const float* qr = qn   + (size_t)b * DIM;
    float s = 0.f;
    for (int k = lane; k < DIM; k += 32) s += er[k] * qr[k];
    for (int off = 16; off; off >>= 1) s += __shfl_down(s, off);
    if (lane == 0) scores[p] = s;
}

// ---------------- zero counts (8*32 ints) --------------------------------------
__global__ void zero_counts(int* __restrict__ c) { c[threadIdx.x] = 0; }

// ---------------- pixel->embedding histogram (bin 30 = uncovered) --------------
__global__ void count_kernel(const int* __restrict__ bboxes,
                             const int* __restrict__ emb_len,
                             int* __restrict__ counts)
{
    const int b = blockIdx.y;
    const int n = blockIdx.x * 256 + threadIdx.x;    // grid.x = 196, exact
    __shared__ int lc[32];
    __shared__ int lbb[NE * 4];
    if (threadIdx.x < 32) lc[threadIdx.x] = 0;
    if (threadIdx.x < NE * 4) lbb[threadIdx.x] = bboxes[b * NE * 4 + threadIdx.x];
    __syncthreads();
    int vE = emb_len[b]; if (vE > NE) vE = NE;
    const int row = n / IMG, col = n % IMG;
    int id = -1;
    for (int e = 0; e < vE; ++e) {
        const int x  = lbb[e*4],   y  = lbb[e*4+1];
        const int x2 = lbb[e*4+2], y2 = lbb[e*4+3];
        if (col >= x && col < x2 && row >= y && row < y2) id = e;  // last valid wins
    }
    atomicAdd(&lc[id < 0 ? 30 : id], 1);
    __syncthreads();
    if (threadIdx.x < 31 && lc[threadIdx.x] != 0)
        atomicAdd(&counts[b * 32 + threadIdx.x], lc[threadIdx.x]);
}

// ---------------- weighted[b,:] = sum_e softmaxweight_e * embn[b,e,:] ----------
__global__ void weighted_kernel(const float* __restrict__ embn,
                                const float* __restrict__ scores,
                                const int* __restrict__ counts,
                                float* __restrict__ outw)
{
    const int b = blockIdx.x;
    const int cneg = counts[b * 32 + 30];
    float m = -INFINITY;
#pragma unroll
    for (int e = 0; e < NE; ++e)
        if (counts[b * 32 + e] > 0) m = fmaxf(m, scores[b * NE + e]);
    if (cneg > 0) m = fmaxf(m, 0.f);
    float Z = (cneg > 0) ? (float)cneg * expf(0.f - m) : 0.f;
    float w[NE];
#pragma unroll
    for (int e = 0; e < NE; ++e) {
        const int c = counts[b * 32 + e];
        const float we = (c > 0) ? (float)c * expf(scores[b * NE + e] - m) : 0.f;
        w[e] = we; Z += we;
    }
    const float invZ = 1.f / Z;
    for (int d = threadIdx.x; d < DIM; d += blockDim.x) {
        float acc = 0.f;
#pragma unroll
        for (int e = 0; e < NE; ++e) acc += w[e] * embn[(size_t)(b * NE + e) * DIM + d];
        outw[b * DIM + d] = acc * invZ;
    }
}

// ---------------- L2 normalize weighted over BATCH dim (in place) --------------
__global__ void bnorm_kernel(float* __restrict__ outw)
{
    const int d = blockIdx.x * blockDim.x + threadIdx.x;
    if (d >= DIM) return;
    float s = 0.f;
#pragma unroll
    for (int b = 0; b < BATCH; ++b) { float v = outw[b * DIM + d]; s += v * v; }
    const float den = fmaxf(sqrtf(s), 1e-12f);
#pragma unroll
    for (int b = 0; b < BATCH; ++b) outw[b * DIM + d] = outw[b * DIM + d] / den;
}

// ---------------- streaming wordgrid writer: [B, 300, HW] ----------------------
// grid (49, 4, 8), block 256; each thread owns 4 consecutive pixels, 75 d's.
// Non-temporal b128 stores: 481 MB write-once stream, keep it out of L2's
// retained set and head straight for HBM (23.3 TB/s roofline, ~21 us floor).
__global__ void wordgrid_kernel(const float* __restrict__ embn,
                                const int* __restrict__ bboxes,
                                const int* __restrict__ emb_len,
                                float* __restrict__ wg)
{
    const int b  = blockIdx.z;
    const int d0 = blockIdx.y * DCH;
    const int nb = blockIdx.x * 1024 + threadIdx.x * 4;
    __shared__ float lemb[NE * DCH];   // 2250 floats
    __shared__ int   lbb[NE * 4];
    const int tid = threadIdx.x;
    if (tid < NE * 4) lbb[tid] = bboxes[b * NE * 4 + tid];
    for (int i = tid; i < NE * DCH; i += 256) {
        const int e = i / DCH, dd = i % DCH;
        lemb[i] = embn[(size_t)(b * NE + e) * DIM + d0 + dd];
    }
    __syncthreads();
    int vE = emb_len[b]; if (vE > NE) vE = NE;
    int idx[4];
#pragma unroll
    for (int j = 0; j < 4; ++j) {
        const int n = nb + j;
        const int row = n / IMG, col = n % IMG;
        int id = -1;
        for (int e = 0; e < vE; ++e) {
            const int x  = lbb[e*4],   y  = lbb[e*4+1];
            const int x2 = lbb[e*4+2], y2 = lbb[e*4+3];
            if (col >= x && col < x2 && row >= y && row < y2) id = e;
        }
        idx[j] = id;
    }
    float* base = wg + (size_t)(b * DIM + d0) * HW + nb;
    for (int dd = 0; dd < DCH; ++dd) {
        v4f v;
        v.x = idx[0] >= 0 ? lemb[idx[0] * DCH + dd] : 0.f;
        v.y = idx[1] >= 0 ? lemb[idx[1] * DCH + dd] : 0.f;
        v.z = idx[2] >= 0 ? lemb[idx[2] * DCH + dd] : 0.f;
        v.w = idx[3] >= 0 ? lemb[idx[3] * DCH + dd] : 0.f;
        __builtin_nontemporal_store(v, reinterpret_cast<v4f*>(base + (size_t)dd * HW));
    }
}

extern "C" void kernel_launch(void* const* d_in, const int* in_sizes, int n_in,
                              void* d_out, int out_size, void* d_ws, size_t ws_size,
                              hipStream_t stream) {
    (void)in_sizes; (void)n_in; (void)out_size; (void)ws_size;
    // setup_inputs order: image, question, question_len, embeddings, bboxes,
    //                     emb_lengths, ocr_W, ocr_b, q_W, q_b
    const float* question    = (const float*)d_in[1];
    const float* embeddings  = (const float*)d_in[3];
    const int*   bboxes      = (const int*)d_in[4];
    const int*   emb_lengths = (const int*)d_in[5];
    const float* ocr_W       = (const float*)d_in[6];
    const float* ocr_b       = (const float*)d_in[7];
    const float* q_W         = (const float*)d_in[8];
    const float* q_b         = (const float*)d_in[9];

    float* out_w  = (float*)d_out;            // [8,300]   weighted
    float* out_q  = out_w + BATCH * DIM;      // [8,1,300] q (normalized in place)
    float* out_wg = out_q + BATCH * DIM;      // [8,300,50176] wordgrid

    float* embn   = (float*)d_ws;                    // 72000 floats
    float* scores = embn + BATCH * NE * DIM;         // 240 floats
    int*   counts = (int*)(scores + 256);            // 8*32 ints

    // 1) linears (WMMA f32 16x16x4) + ReLU
    linear_relu_wmma<<<dim3((15 * NT_N + 7) / 8), 256, 0, stream>>>(
        embeddings, ocr_W, ocr_b, embn, BATCH * NE, 15 * NT_N);
    linear_relu_wmma<<<dim3((NT_N + 7) / 8), 256, 0, stream>>>(
        question, q_W, q_b, out_q, BATCH, NT_N);

    // 2) row L2 norms
    rownorm_kernel<<<BATCH * NE, 128, 0, stream>>>(embn);
    rownorm_kernel<<<BATCH, 128, 0, stream>>>(out_q);

    // 3) per-embedding attention scores
    score_kernel<<<(BATCH * NE) / 8, 256, 0, stream>>>(embn, out_q, scores);

    // 4) pixel coverage histogram
    zero_counts<<<1, BATCH * 32, 0, stream>>>(counts);
    count_kernel<<<dim3(HW / 256, BATCH), 256, 0, stream>>>(bboxes, emb_lengths, counts);

    // 5) closed-form softmax-weighted sum, then batch-dim L2 norm
    weighted_kernel<<<BATCH, 256, 0, stream>>>(embn, scores, counts, out_w);
    bnorm_kernel<<<(DIM + 255) / 256, 256, 0, stream>>>(out_w);

    // 6) stream the 481 MB wordgrid with non-temporal b128 stores
    wordgrid_kernel<<<dim3(HW / 1024, DIM / DCH, BATCH), 256, 0, stream>>>(
        embn, bboxes, emb_lengths, out_wg);
}